// LSTM_19559281066317
// MI455X (gfx1250) — compile-verified
//
#include <hip/hip_runtime.h>

#define T_STEPS 1024
#define BATCH   64
#define HID     256
#define TB      (T_STEPS * BATCH)           // 65536 rows for the big GEMMs
#define BH      (BATCH * HID)               // 16384 elems per timestep slab
static const size_t TBH = (size_t)TB * HID; // 16,777,216 elems per gate slab

typedef float v2f __attribute__((ext_vector_type(2)));
typedef float v8f __attribute__((ext_vector_type(8)));
typedef __attribute__((address_space(3))) float lds_float;

// D = A(16x4 f32) * B(4x16 f32) + C(16x16 f32), wave32 WMMA
__device__ __forceinline__ v8f wmma_f32(v2f a, v2f b, v8f c) {
  return __builtin_amdgcn_wmma_f32_16x16x4_f32(
      /*neg_a=*/false, a, /*neg_b=*/false, b,
      /*c_mod=*/(short)0, c, /*reuse_a=*/false, /*reuse_b=*/false);
}

// ---------------------------------------------------------------- init ------
__global__ void lstm_init(const float* __restrict__ h0,
                          float* __restrict__ h_all,
                          unsigned* __restrict__ sync) {
  int idx = blockIdx.x * blockDim.x + threadIdx.x;
  if (idx < BH) h_all[idx] = h0[idx];          // h_all[0] = h0
  if (idx == 0) { sync[0] = 0u; sync[1] = 0u; } // barrier count / generation
}

// ------------------------------------------------ x-projection (parallel) ---
// gates[g][t*64+b][n] = sum_k x[t,b,k] * Wx_g[k,n] + b_g[n]
// One 16x16 output tile per wave; 8 waves per block; 262144 tiles total.
__global__ void __launch_bounds__(256)
lstm_xproj(const float* __restrict__ x,
           const float* __restrict__ Wxi, const float* __restrict__ bi,
           const float* __restrict__ Wxf, const float* __restrict__ bf,
           const float* __restrict__ Wxo, const float* __restrict__ bo,
           const float* __restrict__ Wxg, const float* __restrict__ bg,
           float* __restrict__ gates) {
  const int lane = threadIdx.x & 31;
  const int wave = threadIdx.x >> 5;
  const int tile = blockIdx.x * 8 + wave;   // 0 .. 262143
  const int nt   = tile & 63;               // gate*16 + coltile
  const int mt   = tile >> 6;               // 0 .. 4095 (rows of TB)
  const int g    = nt >> 4;
  const int ct   = nt & 15;

  // Ternary selects (not a stack array) so address-space inference keeps
  // these on the GLOBAL path instead of FLAT (no DScnt coupling).
  const float* Wg = (g == 0) ? Wxi : (g == 1) ? Wxf : (g == 2) ? Wxo : Wxg;
  const float* Bg = (g == 0) ? bi  : (g == 1) ? bf  : (g == 2) ? bo  : bg;

  const int nCol  = ct * 16 + (lane & 15);
  const int aRow  = mt * 16 + (lane & 15);
  const int khalf = (lane < 16) ? 0 : 2;

  const float bias = Bg[nCol];
  v8f acc = {bias, bias, bias, bias, bias, bias, bias, bias};

  const float* ar = x + (size_t)aRow * HID;
#pragma unroll
  for (int kt = 0; kt < 64; ++kt) {
    const int k0 = kt * 4 + khalf;
    v2f a; a.x = ar[k0];                      a.y = ar[k0 + 1];
    v2f b; b.x = Wg[(size_t)k0 * HID + nCol]; b.y = Wg[(size_t)(k0 + 1) * HID + nCol];
    acc = wmma_f32(a, b, acc);
  }

  float* out = gates + (size_t)g * TBH;
  const int rbase = mt * 16 + ((lane < 16) ? 0 : 8);
#pragma unroll
  for (int r = 0; r < 8; ++r)
    out[(size_t)(rbase + r) * HID + nCol] = acc[r];
}

// ----------------------------------------------------------- grid barrier ---
__device__ __forceinline__ void grid_sync(unsigned* sync, unsigned nblocks) {
  __syncthreads();
  if (threadIdx.x == 0) {
    __threadfence();                          // release h[t] writes
    volatile unsigned* vgen = sync + 1;
    const unsigned g = *vgen;
    if (atomicAdd(sync, 1u) == nblocks - 1u) {
      sync[0] = 0u;
      __threadfence();
      *vgen = g + 1u;
    } else {
      while (*vgen == g) __builtin_amdgcn_s_sleep(2);
    }
    __threadfence();                          // acquire other blocks' h[t]
  }
  __syncthreads();
}

// ------------------------------------------------- recurrent scan (latency) -
// 64 persistent blocks x 128 threads (4 waves). Block owns a 16x16 (batch x H)
// tile; wave w computes gate w. Recurrent weight fragments live in VGPRs.
// h[t-1] tile (16 rows x 256 cols = 16KB, contiguous) is staged into LDS once
// per block via the async-to-LDS engine, then all 4 waves feed WMMA from LDS.
__global__ void __launch_bounds__(128)
lstm_scan(const float* __restrict__ Whi, const float* __restrict__ Whf,
          const float* __restrict__ Who, const float* __restrict__ Whg,
          const float* __restrict__ c0,
          const float* __restrict__ gates,
          float* __restrict__ h_all,
          float* __restrict__ out_tail,   // d_out + TBH: [h_T | c_T]
          unsigned* __restrict__ sync) {
  const int tid   = threadIdx.x;
  const int lane  = tid & 31;
  const int gate  = tid >> 5;               // wave id == gate id
  const int mtile = blockIdx.x & 3;         // batch rows 16*mtile ..
  const int ctile = blockIdx.x >> 2;        // H cols 16*ctile ..
  const int Mb = mtile * 16;
  const int Cb = ctile * 16;
  const int nCol  = Cb + (lane & 15);
  const int khalf = (lane < 16) ? 0 : 2;

  const float* Wh = (gate == 0) ? Whi : (gate == 1) ? Whf : (gate == 2) ? Who : Whg;

  // Preload B fragments for all 64 K-tiles: 128 VGPRs of weights, loaded once.
  v2f wreg[64];
#pragma unroll
  for (int kt = 0; kt < 64; ++kt) {
    const int k0 = kt * 4 + khalf;
    wreg[kt].x = Wh[(size_t)k0 * HID + nCol];
    wreg[kt].y = Wh[(size_t)(k0 + 1) * HID + nCol];
  }

  // Per-thread cell state: elements e = tid, tid+128 of the 16x16 tile.
  float creg[2];
#pragma unroll
  for (int j = 0; j < 2; ++j) {
    const int e = tid + j * 128;
    creg[j] = c0[(size_t)(Mb + (e >> 4)) * HID + Cb + (e & 15)];
  }

  __shared__ float hstage[16 * 256];        // staged h[t-1] tile (16 KB)
  __shared__ float lds[4][256];             // gate pre-activations, 16x16 each

  // Raw LDS byte offset of hstage for the async-to-LDS instruction.
  const unsigned lds_base = (unsigned)(size_t)(lds_float*)hstage;

  const int arowLds = (lane & 15) * HID;    // A-fragment row base in hstage
  const int rhalf   = (lane < 16) ? 0 : 8;

  for (int t = 0; t < T_STEPS; ++t) {
    const float* xg = gates + (size_t)gate * TBH + (size_t)t * BH;

    if (t + 1 < T_STEPS)  // pull next step's x-gate tile toward this WGP
      __builtin_prefetch(xg + BH + (size_t)(Mb + (lane & 15)) * HID + Cb, 0, 1);

    // Async-stage the contiguous 16KB h[t-1] tile: 128 threads x 8 x b128.
    {
      const char* gsrc = (const char*)(h_all + (size_t)t * BH + (size_t)Mb * HID);
#pragma unroll
      for (int i = 0; i < 8; ++i) {
        const unsigned off = (unsigned)(tid + i * 128) * 16u;   // bytes
        const unsigned lds_addr = lds_base + off;
        const unsigned long long ga = (unsigned long long)(size_t)(gsrc + off);
        asm volatile("global_load_async_to_lds_b128 %0, %1, off"
                     :: "v"(lds_addr), "v"(ga) : "memory");
      }
      asm volatile("s_wait_asynccnt 0x0" ::: "memory");
    }
    __syncthreads();

    v8f acc = {0.f, 0.f, 0.f, 0.f, 0.f, 0.f, 0.f, 0.f};
#pragma unroll
    for (int kt = 0; kt < 64; ++kt) {
      const int k0 = kt * 4 + khalf;
      v2f a; a.x = hstage[arowLds + k0]; a.y = hstage[arowLds + k0 + 1];
      acc = wmma_f32(a, wreg[kt], acc);
    }

#pragma unroll
    for (int r = 0; r < 8; ++r) {
      const int row = r + rhalf;            // local row 0..15
      lds[gate][row * 16 + (lane & 15)] =
          acc[r] + xg[(size_t)(Mb + row) * HID + nCol];
    }
    __syncthreads();

    float* hnext = h_all + (size_t)(t + 1) * BH;
#pragma unroll
    for (int j = 0; j < 2; ++j) {
      const int e = tid + j * 128;
      const float iv = 1.f / (1.f + __expf(-lds[0][e]));
      const float fv = 1.f / (1.f + __expf(-lds[1][e]));
      const float ov = 1.f / (1.f + __expf(-lds[2][e]));
      const float gv = tanhf(lds[3][e]);
      const float cv = fv * creg[j] + iv * gv;
      creg[j] = cv;
      hnext[(size_t)(Mb + (e >> 4)) * HID + Cb + (e & 15)] = ov * tanhf(cv);
    }
    grid_sync(sync, 64u);  // release h[t+1]; also guards LDS reuse
  }

  // Tail outputs: h_T and c_T
#pragma unroll
  for (int j = 0; j < 2; ++j) {
    const int e = tid + j * 128;
    const size_t idx = (size_t)(Mb + (e >> 4)) * HID + Cb + (e & 15);
    out_tail[idx]      = h_all[(size_t)T_STEPS * BH + idx];
    out_tail[BH + idx] = creg[j];
  }
}

// ------------------------------------------------ y-projection (parallel) ---
__global__ void __launch_bounds__(256)
lstm_yproj(const float* __restrict__ h_all,
           const float* __restrict__ Why, const float* __restrict__ by,
           float* __restrict__ y) {
  const int lane = threadIdx.x & 31;
  const int wave = threadIdx.x >> 5;
  const int tile = blockIdx.x * 8 + wave;   // 0 .. 65535
  const int ct = tile & 15;
  const int mt = tile >> 4;
  const int nCol  = ct * 16 + (lane & 15);
  const int aRow  = mt * 16 + (lane & 15);
  const int khalf = (lane < 16) ? 0 : 2;

  const float bias = by[nCol];
  v8f acc = {bias, bias, bias, bias, bias, bias, bias, bias};

  const float* ar = h_all + BH /* skip h0 */ + (size_t)aRow * HID;
#pragma unroll
  for (int kt = 0; kt < 64; ++kt) {
    const int k0 = kt * 4 + khalf;
    v2f a; a.x = ar[k0];                       a.y = ar[k0 + 1];
    v2f b; b.x = Why[(size_t)k0 * HID + nCol]; b.y = Why[(size_t)(k0 + 1) * HID + nCol];
    acc = wmma_f32(a, b, acc);
  }
  const int rbase = mt * 16 + ((lane < 16) ? 0 : 8);
#pragma unroll
  for (int r = 0; r < 8; ++r)
    y[(size_t)(rbase + r) * HID + nCol] = acc[r];
}

// --------------------------------------------------------------- launcher ---
extern "C" void kernel_launch(void* const* d_in, const int* in_sizes, int n_in,
                              void* d_out, int out_size, void* d_ws, size_t ws_size,
                              hipStream_t stream) {
  const float* x   = (const float*)d_in[0];
  const float* h0  = (const float*)d_in[1];
  const float* c0  = (const float*)d_in[2];
  const float* Wxi = (const float*)d_in[3];
  const float* Whi = (const float*)d_in[4];
  const float* bi  = (const float*)d_in[5];
  const float* Wxf = (const float*)d_in[6];
  const float* Whf = (const float*)d_in[7];
  const float* bf  = (const float*)d_in[8];
  const float* Wxo = (const float*)d_in[9];
  const float* Who = (const float*)d_in[10];
  const float* bo  = (const float*)d_in[11];
  const float* Wxg = (const float*)d_in[12];
  const float* Whg = (const float*)d_in[13];
  const float* bg  = (const float*)d_in[14];
  const float* Why = (const float*)d_in[15];
  const float* by  = (const float*)d_in[16];

  // Workspace layout: [gates: 4*TBH f32][h_all: (T+1)*BH f32][sync: 2 u32]
  float*    gates = (float*)d_ws;
  float*    h_all = gates + 4 * TBH;
  unsigned* sync  = (unsigned*)(h_all + (size_t)(T_STEPS + 1) * BH);

  float* y        = (float*)d_out;       // [T,B,256]
  float* out_tail = y + TBH;             // [h_T | c_T]

  lstm_init <<<64,    256, 0, stream>>>(h0, h_all, sync);
  lstm_xproj<<<32768, 256, 0, stream>>>(x, Wxi, bi, Wxf, bf, Wxo, bo, Wxg, bg, gates);
  lstm_scan <<<64,    128, 0, stream>>>(Whi, Whf, Who, Whg, c0, gates, h_all,
                                        out_tail, sync);
  lstm_yproj<<<8192,  256, 0, stream>>>(h_all, Why, by, y);
}